// MyRGCNConvOpt1_35536559407733
// MI455X (gfx1250) — compile-verified
//
#include <hip/hip_runtime.h>

typedef __attribute__((ext_vector_type(2))) float v2f;
typedef __attribute__((ext_vector_type(8))) float v8f;

#define TILE_M 32              // nodes per block = 2 WMMA M-tiles (register-blocked)
#define NREL 8
#define CIN 128
#define CHID 128
#define AGG_STRIDE (CIN + 4)   // bank-conflict-free A-fragment LDS reads
#define THREADS 256            // 8 wave32 waves
#define NWAVE (THREADS / 32)

__global__ __launch_bounds__(THREADS)
void rgcn_fused_kernel(const float* __restrict__ x,
                       const float* __restrict__ W,     // [NREL][CIN][CHID]
                       const int*   __restrict__ ptr,   // [n_node+1] CSR
                       const int*   __restrict__ idx,   // [n_edge]
                       const int*   __restrict__ rel,   // [n_edge]
                       float*       __restrict__ out,   // [n_node][CHID]
                       int n_node) {
  __shared__ float aggS[NREL * TILE_M * AGG_STRIDE];    // ~135 KB -> 2 blocks / WGP
  __shared__ int   ptrS[TILE_M + 1];

  const int tid  = threadIdx.x;
  const int lane = tid & 31;
  const int wv   = tid >> 5;            // 0..7
  const int n0   = blockIdx.x * TILE_M;

  // ---- zero per-(relation,node) aggregation tiles ----
  for (int i = tid; i < NREL * TILE_M * AGG_STRIDE; i += THREADS) aggS[i] = 0.0f;
  if (tid <= TILE_M) {
    int nn = n0 + tid; if (nn > n_node) nn = n_node;
    ptrS[tid] = ptr[nn];
  }
  __syncthreads();

  // ---- phase 1: aggregate x rows per (relation, local node) ----
  // Edges of this 32-node tile are contiguous [ptrS[0], ptrS[32]).
  // One edge per wave per step (wave-uniform control), lane covers 4 channels.
  {
    const int estart = ptrS[0];
    const int eend   = ptrS[TILE_M];
    for (int e = estart + wv; e < eend; e += NWAVE) {
      // local node: largest ln with ptrS[ln] <= e (wave-uniform binary search)
      int lo = 0, hi = TILE_M;
      while (hi - lo > 1) { int mid = (lo + hi) >> 1; if (ptrS[mid] <= e) lo = mid; else hi = mid; }
      const int r  = rel[e];
      const int id = idx[e];
      // prefetch next edge's x row toward L0/L2 (global_prefetch_b8)
      const int en = e + NWAVE;
      if (en < eend)
        __builtin_prefetch(x + (size_t)idx[en] * CIN + lane * 4, 0, 3);
      const float4 xv = ((const float4*)(x + (size_t)id * CIN))[lane];  // L2-resident gather
      float* dst = &aggS[(r * TILE_M + lo) * AGG_STRIDE + lane * 4];
      atomicAdd(dst + 0, xv.x);   // ds_add_f32
      atomicAdd(dst + 1, xv.y);
      atomicAdd(dst + 2, xv.z);
      atomicAdd(dst + 3, xv.w);
    }
  }
  __syncthreads();

  // ---- phase 2: out_tile[32 x 128] = sum_r aggS[r](32x128) @ W[r](128x128) ----
  // Each wave owns one 16-wide N slice; one B fragment feeds TWO M-tiles (c0,c1),
  // halving L2 weight traffic per WMMA. f32 WMMA == exact fp32 semantics.
  const int h     = lane >> 4;      // lane half selects K pair / M half
  const int m     = lane & 15;      // A row / B,C column
  const int nbase = wv * 16;        // this wave's output-channel slice
  v8f c0 = {}, c1 = {};
  for (int r = 0; r < NREL; ++r) {
    const float* __restrict__ wr = W + (size_t)r * CIN * CHID;
    const float* ag = &aggS[(r * TILE_M) * AGG_STRIDE];
#pragma unroll 4
    for (int k = 0; k < CIN; k += 4) {
      const int krow = k + 2 * h;
      // B fragment 4x16: lane(n=nbase+m), vgpr j -> element (k + 2h + j, n)
      v2f b;
      b.x = wr[(size_t)krow * CHID + nbase + m];
      b.y = wr[(size_t)(krow + 1) * CHID + nbase + m];
      // A fragments for M-tiles 0 and 1: lane(m), vgpr j -> (m [+16], k + 2h + j)
      v2f a0;
      a0.x = ag[m * AGG_STRIDE + krow];
      a0.y = ag[m * AGG_STRIDE + krow + 1];
      v2f a1;
      a1.x = ag[(m + 16) * AGG_STRIDE + krow];
      a1.y = ag[(m + 16) * AGG_STRIDE + krow + 1];
      c0 = __builtin_amdgcn_wmma_f32_16x16x4_f32(false, a0, false, b, (short)0, c0, false, false);
      c1 = __builtin_amdgcn_wmma_f32_16x16x4_f32(false, a1, false, b, (short)0, c1, false, false);
    }
  }

  // ---- D fragment stores: vgpr v -> row M = v + 8h (per tile), col = nbase + m ----
  const int col = nbase + m;
#pragma unroll
  for (int v = 0; v < 8; ++v) {
    const int node0 = n0 + v + 8 * h;
    if (node0 < n_node) out[(size_t)node0 * CHID + col] = c0[v];
  }
#pragma unroll
  for (int v = 0; v < 8; ++v) {
    const int node1 = n0 + 16 + v + 8 * h;
    if (node1 < n_node) out[(size_t)node1 * CHID + col] = c1[v];
  }
}

extern "C" void kernel_launch(void* const* d_in, const int* in_sizes, int n_in,
                              void* d_out, int out_size, void* d_ws, size_t ws_size,
                              hipStream_t stream) {
  const float* x   = (const float*)d_in[0];
  const float* W   = (const float*)d_in[1];
  const int*   ptr = (const int*)d_in[2];
  const int*   idx = (const int*)d_in[3];
  const int*   rel = (const int*)d_in[4];
  float*       out = (float*)d_out;

  const int n_node = in_sizes[2] - 1;                 // ptr has n_node+1 entries
  const int grid   = (n_node + TILE_M - 1) / TILE_M;  // 1563 blocks for 50000 nodes

  rgcn_fused_kernel<<<grid, THREADS, 0, stream>>>(x, W, ptr, idx, rel, out, n_node);
}